// BatchedCauchyKernel_CONCERT_3891240370724
// MI455X (gfx1250) — compile-verified
//
#include <hip/hip_runtime.h>
#include <hip/hip_bf16.h>

typedef float v2f __attribute__((ext_vector_type(2)));
typedef float v8f __attribute__((ext_vector_type(8)));

#define N_ROWS 8192
#define M_COLS 4096
#define K_DIM  512
#define S_DIM  16

#define BM 128
#define BN 128
#define BK 16
#define KT (K_DIM / BK)
#define PITCH 20   // BK+4: conflict-free b64 fragment reads, b128 stores stay 16B aligned

// ---- CDNA5 async global->LDS path (guarded; falls back to reg staging) ----
#if defined(__gfx1250__) && __has_builtin(__builtin_amdgcn_global_load_async_to_lds_b128)
#define USE_ASYNC_LDS 1
typedef int v4i_ __attribute__((vector_size(16)));            // matches builtin param pointee
typedef __attribute__((address_space(1))) v4i_ gv4i;          // global (prints as __device__)
typedef __attribute__((address_space(3))) v4i_ lv4i;          // LDS (prints as __shared__)
#if __has_builtin(__builtin_amdgcn_s_wait_asynccnt)
#define WAIT_ASYNC(n) __builtin_amdgcn_s_wait_asynccnt(n)
#else
#define WAIT_ASYNC(n) asm volatile("s_wait_asynccnt " #n ::: "memory")
#endif
#else
#define USE_ASYNC_LDS 0
#define WAIT_ASYNC(n) ((void)0)
#endif

// ---------------------------------------------------------------------------
// Per-row stats: sum of squares and clipped softplus lengthscale.
// One wave per row; 8 rows per 256-thread block.
// ---------------------------------------------------------------------------
__global__ __launch_bounds__(256) void row_stats_kernel(
    const float* __restrict__ v, const float* __restrict__ samp,
    const float* __restrict__ scale, float2* __restrict__ out, int rows)
{
  const int wid  = threadIdx.x >> 5;
  const int lane = threadIdx.x & 31;
  const int row  = blockIdx.x * 8 + wid;
  if (row >= rows) return;

  const float* vr = v + (size_t)row * K_DIM;
  float s = 0.f;
  #pragma unroll
  for (int i = 0; i < K_DIM / 32; ++i) {
    float t = vr[lane + i * 32];
    s += t * t;
  }
  float z = (lane < S_DIM) ? samp[row * S_DIM + lane] * scale[lane] : 0.f;
  #pragma unroll
  for (int off = 16; off > 0; off >>= 1) {
    s += __shfl_xor(s, off, 32);
    z += __shfl_xor(z, off, 32);
  }
  if (lane == 0) {
    float sp = (z > 20.f) ? z : log1pf(__expf(z));      // softplus, overflow-safe
    sp = fminf(fmaxf(sp, 1e-10f), 10000.f);             // clip
    out[row] = make_float2(s, sp);
  }
}

// ---------------------------------------------------------------------------
// Tiled fp32 GEMM on V_WMMA_F32_16X16X4_F32 with fused Cauchy + cutoff epilogue.
// Block tile 128x128, 8 waves (4 along M, 2 along N), wave tile 32x64.
// Double-buffered LDS, next tile copied with GLOBAL_LOAD_ASYNC_TO_LDS_B128.
// ---------------------------------------------------------------------------
__global__ __launch_bounds__(256) void cauchy_wmma_kernel(
    const float* __restrict__ x, const float* __restrict__ y,
    const float2* __restrict__ statsX, const float2* __restrict__ statsY,
    const float* __restrict__ cutoffP, const float* __restrict__ phiP,
    float* __restrict__ out)
{
  __shared__ float  As[2][BM * PITCH];
  __shared__ float  Bs[2][BN * PITCH];
  __shared__ float2 RowSt[BM];
  __shared__ float2 ColSt[BN];

  const int tid  = threadIdx.x;
  const int lane = tid & 31;
  const int wid  = tid >> 5;
  const int wm   = wid & 3;       // wave row (0..3) -> 32 rows each
  const int wn   = wid >> 2;      // wave col (0..1) -> 64 cols each
  const int row0 = blockIdx.y * BM;
  const int col0 = blockIdx.x * BN;

  if (tid < BM) RowSt[tid]      = statsX[row0 + tid];
  else          ColSt[tid - BM] = statsY[col0 + tid - BM];

  const float cutv = fminf(fmaxf(cutoffP[0], 0.f), 1000.f);
  const float phiv = phiP[0];

  // load-slot assignment: 2 x 16B per matrix per K-tile per thread
  const int f0 = tid, f1 = tid + 256;
  const int r0 = f0 >> 2, c0 = f0 & 3;   // BK/4 = 4 float4 per row
  const int r1 = f1 >> 2, c1 = f1 & 3;

  const float* xp0 = x + (size_t)(row0 + r0) * K_DIM + c0 * 4;
  const float* xp1 = x + (size_t)(row0 + r1) * K_DIM + c1 * 4;
  const float* yp0 = y + (size_t)(col0 + r0) * K_DIM + c0 * 4;
  const float* yp1 = y + (size_t)(col0 + r1) * K_DIM + c1 * 4;
  const int la0 = r0 * PITCH + c0 * 4;
  const int la1 = r1 * PITCH + c1 * 4;

  v8f acc[2][4];
  #pragma unroll
  for (int i = 0; i < 2; ++i)
    #pragma unroll
    for (int j = 0; j < 4; ++j)
      #pragma unroll
      for (int e = 0; e < 8; ++e)
        acc[i][j][e] = 0.f;

  // stage one K-tile into LDS buffer `buf` (4 x 16B per thread)
  auto issue_tile = [&](int kt, int buf) {
    const int ko = kt * BK;
#if USE_ASYNC_LDS
    __builtin_amdgcn_global_load_async_to_lds_b128(
        (gv4i*)(xp0 + ko), (lv4i*)&As[buf][la0], 0, 0);
    __builtin_amdgcn_global_load_async_to_lds_b128(
        (gv4i*)(xp1 + ko), (lv4i*)&As[buf][la1], 0, 0);
    __builtin_amdgcn_global_load_async_to_lds_b128(
        (gv4i*)(yp0 + ko), (lv4i*)&Bs[buf][la0], 0, 0);
    __builtin_amdgcn_global_load_async_to_lds_b128(
        (gv4i*)(yp1 + ko), (lv4i*)&Bs[buf][la1], 0, 0);
#else
    float4 ta0 = *(const float4*)(xp0 + ko);
    float4 ta1 = *(const float4*)(xp1 + ko);
    float4 tb0 = *(const float4*)(yp0 + ko);
    float4 tb1 = *(const float4*)(yp1 + ko);
    *(float4*)(&As[buf][la0]) = ta0;
    *(float4*)(&As[buf][la1]) = ta1;
    *(float4*)(&Bs[buf][la0]) = tb0;
    *(float4*)(&Bs[buf][la1]) = tb1;
#endif
  };

  const int lr    = lane & 15;
  const int kb_hi = (lane >> 4) << 1;   // lanes 16-31 hold K+2, K+3 of each frag

  issue_tile(0, 0);                     // prologue: tile 0 -> buffer 0

  for (int kt = 0; kt < KT; ++kt) {
    const int p = kt & 1;
    if (kt + 1 < KT) {
      issue_tile(kt + 1, p ^ 1);        // next tile in flight while we compute
      WAIT_ASYNC(4);                    // in-order: current tile's 4 copies done
    } else {
      WAIT_ASYNC(0);
    }
    __syncthreads();                    // every wave waited on its own copies

    #pragma unroll
    for (int kk = 0; kk < BK; kk += 4) {
      const int kb = kk + kb_hi;
      v2f a0 = *(const v2f*)(&As[p][(32 * wm +      lr) * PITCH + kb]);
      v2f a1 = *(const v2f*)(&As[p][(32 * wm + 16 + lr) * PITCH + kb]);
      #pragma unroll
      for (int j = 0; j < 4; ++j) {
        v2f b = *(const v2f*)(&Bs[p][(64 * wn + 16 * j + lr) * PITCH + kb]);
        acc[0][j] = __builtin_amdgcn_wmma_f32_16x16x4_f32(
            false, a0, false, b, (short)0, acc[0][j], false, false);
        acc[1][j] = __builtin_amdgcn_wmma_f32_16x16x4_f32(
            false, a1, false, b, (short)0, acc[1][j], false, false);
      }
    }
    __syncthreads();                    // all reads of buffer p done before reuse
  }

  // fused epilogue: d = |x|^2 + |y|^2 - 2*dot; res = s/(s+d); res *= sigmoid(phi*(res-cut))
  const int rhi = (lane >> 4) << 3;     // C layout: lanes 16-31 hold M = r + 8
  #pragma unroll
  for (int i = 0; i < 2; ++i) {
    #pragma unroll
    for (int j = 0; j < 4; ++j) {
      #pragma unroll
      for (int r = 0; r < 8; ++r) {
        const int lrow = 32 * wm + 16 * i + r + rhi;
        const int lcol = 64 * wn + 16 * j + lr;
        const float dotv = acc[i][j][r];
        const float2 rs = RowSt[lrow];
        const float2 cs = ColSt[lcol];
        const float d   = rs.x + cs.x - 2.f * dotv;
        const float sxy = sqrtf(rs.y * cs.y);
        const float res = sxy / (sxy + d);
        const float msk = 1.f / (1.f + __expf(-phiv * (res - cutv)));
        out[(size_t)(row0 + lrow) * M_COLS + (col0 + lcol)] = res * msk;
      }
    }
  }
}

// ---------------------------------------------------------------------------
extern "C" void kernel_launch(void* const* d_in, const int* in_sizes, int n_in,
                              void* d_out, int out_size, void* d_ws, size_t ws_size,
                              hipStream_t stream) {
  const float* x        = (const float*)d_in[0];
  const float* y        = (const float*)d_in[1];
  const float* sample_x = (const float*)d_in[2];
  const float* sample_y = (const float*)d_in[3];
  const float* scale    = (const float*)d_in[4];
  const float* cutoff   = (const float*)d_in[5];
  const float* phi      = (const float*)d_in[6];
  float* out            = (float*)d_out;

  float2* statsX = (float2*)d_ws;              // 8192 float2
  float2* statsY = statsX + N_ROWS;            // 4096 float2  (96 KB total)

  row_stats_kernel<<<N_ROWS / 8, 256, 0, stream>>>(x, sample_x, scale, statsX, N_ROWS);
  row_stats_kernel<<<M_COLS / 8, 256, 0, stream>>>(y, sample_y, scale, statsY, M_COLS);

  dim3 grid(M_COLS / BN, N_ROWS / BM);         // (32, 64)
  cauchy_wmma_kernel<<<grid, 256, 0, stream>>>(x, y, statsX, statsY, cutoff, phi, out);
}